// TokenChoiceTopKRouter_66915590472169
// MI455X (gfx1250) — compile-verified
//
#include <hip/hip_runtime.h>

typedef __attribute__((ext_vector_type(2))) float v2f;
typedef __attribute__((ext_vector_type(8))) float v8f;

#define NUM_EXPERTS 64
#define TOP_K       8
#define TOKENS      16384
#define DIM         4096
#define KC          64               // K-chunk staged in LDS (double buffered)
#define NCH         (DIM / KC)       // 64 chunks
#define LDB         (KC + 4)         // 68: 16B-aligned rows, conflict-free reads
#define BUF_FLOATS  (NUM_EXPERTS * LDB)
#define LDL         66               // padded stride for logits scratch
#define ROUTE_SCALE 2.5f

// d_out layout (all float): [top_scores 16384*8][indices 16384*8][counts 64]
__global__ void zero_counts_kernel(float* __restrict__ out) {
  out[2 * (size_t)TOKENS * TOP_K + threadIdx.x] = 0.0f;
}

__global__ __launch_bounds__(256) void router_kernel(
    const float* __restrict__ x,      // (16384, 4096)
    const float* __restrict__ gw,     // (64, 4096)
    const float* __restrict__ bias,   // (64,)
    float* __restrict__ out)
{
  __shared__ union {
    float B[2][BUF_FLOATS];           // 2*17408 B = 34816 B (gate_w chunks)
    float L[8 * 16 * LDL];            // 33792 B (logit tiles, reused)
  } shm;
  __shared__ float biasS[NUM_EXPERTS];
  __shared__ int   hist[NUM_EXPERTS];

  const int tid  = threadIdx.x;
  const int wave = tid >> 5;
  const int lane = tid & 31;
  const int l16  = lane & 15;
  const int hi   = lane >> 4;        // lane group: 0 -> K {0,1}, 1 -> K {2,3}

  if (tid < NUM_EXPERTS) { biasS[tid] = bias[tid]; hist[tid] = 0; }

  const int tokBase = (blockIdx.x * 8 + wave) * 16;
  const float* xrow = x + (size_t)(tokBase + l16) * DIM;

  // Uniform 64-bit base of gate_w for the saddr form of the async load.
  const unsigned long long gwBase = (unsigned long long)(size_t)gw;
  // LDS byte offset of the staging buffers (generic->LDS: low 32 bits).
  const unsigned ldsBBase = (unsigned)(size_t)(&shm.B[0][0]);

  // Stage gate_w[:, kb:kb+KC) into LDS buffer `bufsel` via the async engine.
  // 64*KC*4 B = 16 KB -> 1024 b128 transfers -> 4 per thread (ASYNCcnt path).
  auto stage = [&](int kb, int bufsel) {
    const unsigned base = ldsBBase + (unsigned)bufsel * (BUF_FLOATS * 4u);
#pragma unroll
    for (int j = 0; j < 4; ++j) {
      const int i  = tid + 256 * j;          // 0..1023
      const int e  = i >> 4;                 // i / (KC/4 = 16)
      const int kv = (i & 15) * 4;           // float index within chunk row
      const unsigned ldsAddr = base + (unsigned)(e * LDB + kv) * 4u;
      const unsigned gOff    = (unsigned)((e * DIM + kb + kv) * 4);
      asm volatile("global_load_async_to_lds_b128 %0, %1, %2"
                   :: "v"(ldsAddr), "v"(gOff), "s"(gwBase)
                   : "memory");
    }
  };

  v8f acc0 = {}, acc1 = {}, acc2 = {}, acc3 = {};

  stage(0, 0);                         // prologue: chunk 0 in flight

  for (int c = 0; c < NCH; ++c) {
    const int kb = c * KC;

    if (c + 1 < NCH) {
      // Issue next chunk into the other buffer, then wait only for the
      // previous chunk's 4 transfers (ASYNCcnt completes in order).
      stage(kb + KC, (c + 1) & 1);
      asm volatile("s_wait_asynccnt 0x4" ::: "memory");
    } else {
      asm volatile("s_wait_asynccnt 0x0" ::: "memory");
    }
    __syncthreads();                   // chunk c visible to all waves

    // Keep the HBM x-stream hot across the chunk boundary.
    if (kb + KC < DIM)
      __builtin_prefetch(xrow + kb + KC, 0, 3);

    const float* Bc = shm.B[c & 1];
#pragma unroll 4
    for (int k0 = 0; k0 < KC; k0 += 4) {
      const int ko = k0 + 2 * hi;
      // A fragment (16x4 f32): one aligned float2 per lane from global.
      const v2f a = *(const v2f*)(xrow + kb + ko);
      // B fragments (4x16 f32) for the 4 expert tiles, from LDS.
      const v2f b0 = *(const v2f*)(&Bc[( 0 + l16) * LDB + ko]);
      const v2f b1 = *(const v2f*)(&Bc[(16 + l16) * LDB + ko]);
      const v2f b2 = *(const v2f*)(&Bc[(32 + l16) * LDB + ko]);
      const v2f b3 = *(const v2f*)(&Bc[(48 + l16) * LDB + ko]);
      acc0 = __builtin_amdgcn_wmma_f32_16x16x4_f32(false, a, false, b0, (short)0, acc0, false, false);
      acc1 = __builtin_amdgcn_wmma_f32_16x16x4_f32(false, a, false, b1, (short)0, acc1, false, false);
      acc2 = __builtin_amdgcn_wmma_f32_16x16x4_f32(false, a, false, b2, (short)0, acc2, false, false);
      acc3 = __builtin_amdgcn_wmma_f32_16x16x4_f32(false, a, false, b3, (short)0, acc3, false, false);
    }
    __syncthreads();                   // all waves done with buf[c&1]
  }

  // Spill accumulators: per-wave 16-token x 64-expert logit tile.
  // C/D layout: VGPR r, lanes 0-15 -> M=r, lanes 16-31 -> M=8+r; N = lane%16.
  float* Lw = &shm.L[wave * 16 * LDL];
#pragma unroll
  for (int r = 0; r < 8; ++r) {
    const int m = r + 8 * hi;
    float* d = Lw + m * LDL;
    d[ 0 + l16] = acc0[r];
    d[16 + l16] = acc1[r];
    d[32 + l16] = acc2[r];
    d[48 + l16] = acc3[r];
  }
  __syncthreads();

  if (lane < 16) {
    const int t = tokBase + lane;
    float* row = Lw + lane * LDL;
    // fp32 sigmoid scoring
    for (int e = 0; e < NUM_EXPERTS; ++e)
      row[e] = 1.0f / (1.0f + __expf(-row[e]));

    unsigned long long used = 0ull;
    float sc[TOP_K];
    int   si[TOP_K];
    float sum = 0.0f;
#pragma unroll
    for (int k = 0; k < TOP_K; ++k) {
      float best = -3.402823466e38f;
      int   bi   = 0;
      for (int e = 0; e < NUM_EXPERTS; ++e) {
        if ((used >> e) & 1ull) continue;
        const float v = row[e] + biasS[e];   // bias affects selection only
        if (v > best) { best = v; bi = e; }  // strict > : lowest index wins ties
      }
      used |= (1ull << bi);
      sc[k] = row[bi];                       // gathered score excludes bias
      si[k] = bi;
      sum  += sc[k];
    }
    const float scale = ROUTE_SCALE / (sum + 1e-20f);
    float* outs = out + (size_t)t * TOP_K;
    float* outi = out + (size_t)TOKENS * TOP_K + (size_t)t * TOP_K;
#pragma unroll
    for (int k = 0; k < TOP_K; ++k) {
      outs[k] = sc[k] * scale;
      outi[k] = (float)si[k];
      atomicAdd(&hist[si[k]], 1);
    }
  }
  __syncthreads();
  if (tid < NUM_EXPERTS)
    atomicAdd(out + 2 * (size_t)TOKENS * TOP_K + tid, (float)hist[tid]);
}

extern "C" void kernel_launch(void* const* d_in, const int* in_sizes, int n_in,
                              void* d_out, int out_size, void* d_ws, size_t ws_size,
                              hipStream_t stream) {
  const float* x    = (const float*)d_in[0];
  const float* gw   = (const float*)d_in[1];
  const float* bias = (const float*)d_in[2];
  float* out = (float*)d_out;

  zero_counts_kernel<<<1, NUM_EXPERTS, 0, stream>>>(out);
  router_kernel<<<TOKENS / 128, 256, 0, stream>>>(x, gw, bias, out);
}